// GATNet_71536975282838
// MI455X (gfx1250) — compile-verified
//
#include <hip/hip_runtime.h>
#include <math.h>

typedef float v2f __attribute__((ext_vector_type(2)));
typedef float v8f __attribute__((ext_vector_type(8)));

#define NEG_SLOPE 0.2f
// encoded -FLT_MAX for ordered-uint float max trick
#define ENC_NEGINF 0x00800000u

__device__ __forceinline__ unsigned encf(float f) {
    unsigned u = __float_as_uint(f);
    return (u & 0x80000000u) ? ~u : (u | 0x80000000u);
}
__device__ __forceinline__ float decf(unsigned u) {
    return (u & 0x80000000u) ? __uint_as_float(u & 0x7fffffffu)
                             : __uint_as_float(~u);
}
__device__ __forceinline__ float lrelu(float x) { return x > 0.f ? x : NEG_SLOPE * x; }

__device__ __forceinline__ void edge_sd(const long long* __restrict__ ei,
                                        long long E, long long e, int& s, int& d) {
    if (e < E) { s = (int)ei[e]; d = (int)ei[E + e]; }
    else       { s = d = (int)(e - E); }            // appended self-loops
}

// ---------------- init: zero accumulators, set running-max to -inf ----------------
__global__ void k_init(float* out1, float* s1, float* s2, float* dout,
                       unsigned* m1, unsigned* m2, int N) {
    int i = blockIdx.x * blockDim.x + threadIdx.x;
    int t = i;
    if (t < N * 32) { out1[t] = 0.f; return; }          t -= N * 32;
    if (t < N * 2)  { s1[t] = 0.f;  return; }           t -= N * 2;
    if (t < N)      { s2[t] = 0.f;  return; }           t -= N;
    if (t < N * 2)  { dout[t] = 0.f; return; }          t -= N * 2;
    if (t < N * 2)  { m1[t] = ENC_NEGINF; return; }     t -= N * 2;
    if (t < N)      { m2[t] = ENC_NEGINF; return; }
}

// ---------------- layer 1 projection: h1 = x @ W1 via V_WMMA_F32_16X16X4_F32 ------
// Also emits attention pre-logits als1/ald1 = x . (W1_blk @ a) per head.
__global__ void k_proj1_wmma(const float* __restrict__ x, const float* __restrict__ W1,
                             const float* __restrict__ a_src1, const float* __restrict__ a_dst1,
                             float* __restrict__ h1, float* __restrict__ als1,
                             float* __restrict__ ald1, int nTiles, int N) {
    int lane = threadIdx.x & 31;
    int wave = threadIdx.x >> 5;
    int tile = blockIdx.x * (blockDim.x >> 5) + wave;
    if (tile >= nTiles) return;                   // uniform per wave: EXEC stays full
    int base = tile * 16;
    int nn   = lane & 15;                          // row (A) / col (B,C)
    int half = lane >> 4;                          // K pair select
    int k0   = half * 2;
    int arow = min(base + nn, N - 1);

    // A 16x4 f32: VGPR0 = K{0|2}, VGPR1 = K{1|3}
    v2f a; a.x = x[arow * 4 + k0]; a.y = x[arow * 4 + k0 + 1];
    // B 4x16: rows striped across lanes, mirroring A's K split
    v2f b0, b1v;
    b0.x  = W1[(k0    ) * 32 + nn];      b0.y  = W1[(k0 + 1) * 32 + nn];
    b1v.x = W1[(k0    ) * 32 + 16 + nn]; b1v.y = W1[(k0 + 1) * 32 + 16 + nn];

    v8f c0 = {}; v8f c1 = {};
    c0 = __builtin_amdgcn_wmma_f32_16x16x4_f32(false, a, false, b0,  (short)0, c0, false, false);
    c1 = __builtin_amdgcn_wmma_f32_16x16x4_f32(false, a, false, b1v, (short)0, c1, false, false);

#pragma unroll
    for (int r = 0; r < 8; ++r) {                  // C: lanes<16 -> M=r, lanes>=16 -> M=8+r
        int row = base + half * 8 + r;
        if (row < N) {
            h1[(size_t)row * 32 + nn]      = c0[r];
            h1[(size_t)row * 32 + 16 + nn] = c1[r];
        }
    }

    // attention pre-logits: lane -> (node = lane>>1, head = lane&1)
    int node = lane >> 1, hd = lane & 1;
    int nrow = min(base + node, N - 1);
    float als = 0.f, ald = 0.f;
#pragma unroll
    for (int k = 0; k < 4; ++k) {
        float ws = 0.f, wd = 0.f;
#pragma unroll
        for (int cc = 0; cc < 16; ++cc) {
            float w = W1[k * 32 + hd * 16 + cc];
            ws += w * a_src1[hd * 16 + cc];
            wd += w * a_dst1[hd * 16 + cc];
        }
        float xv = x[nrow * 4 + k];
        als += xv * ws; ald += xv * wd;
    }
    if (base + node < N) {
        als1[(base + node) * 2 + hd] = als;
        ald1[(base + node) * 2 + hd] = ald;
    }
}

// ---------------- layer 1 edge pass A: segment max of leaky-relu logits -----------
__global__ void k_l1_max(const long long* __restrict__ ei, long long E, long long ET,
                         const float* __restrict__ als1, const float* __restrict__ ald1,
                         unsigned* __restrict__ m1) {
    long long t = (long long)blockIdx.x * blockDim.x + threadIdx.x;
    if (t >= ET * 2) return;
    long long e = t >> 1;
    int hd = (int)(t & 1);
    int s, d; edge_sd(ei, E, e, s, d);
    float es = lrelu(als1[s * 2 + hd] + ald1[d * 2 + hd]);
    atomicMax(&m1[d * 2 + hd], encf(es));
}

// ---------------- layer 1 edge pass B: out1[dst] += ex*h1[src]; s1[dst] += ex -----
__global__ void k_l1_acc(const long long* __restrict__ ei, long long E, long long ET,
                         const float* __restrict__ als1, const float* __restrict__ ald1,
                         const unsigned* __restrict__ m1, const float* __restrict__ h1,
                         float* __restrict__ out1, float* __restrict__ s1) {
    long long t = (long long)blockIdx.x * blockDim.x + threadIdx.x;
    if (t >= ET * 8) return;
    long long e = t >> 3;
    int cbase = (int)(t & 7) * 4;                   // 4 channels per lane, 8 lanes per edge
    int hd = cbase >> 4;
    int s, d; edge_sd(ei, E, e, s, d);
    float es = lrelu(als1[s * 2 + hd] + ald1[d * 2 + hd]);
    float ex = __expf(es - decf(m1[d * 2 + hd]));
    float4 hv = *(const float4*)(h1 + (size_t)s * 32 + cbase);
    float* o = out1 + (size_t)d * 32 + cbase;
    atomicAdd(o + 0, ex * hv.x);
    atomicAdd(o + 1, ex * hv.y);
    atomicAdd(o + 2, ex * hv.z);
    atomicAdd(o + 3, ex * hv.w);
    if ((cbase & 15) == 0) atomicAdd(&s1[d * 2 + hd], ex);
}

// ---------------- layer 1 finalize: h = elu(out1 / s1 + b1), reuse h1 buffer ------
__global__ void k_fin1(const float* __restrict__ out1, const float* __restrict__ s1,
                       const float* __restrict__ b1, float* __restrict__ hpost, int N) {
    int i = blockIdx.x * blockDim.x + threadIdx.x;
    if (i >= N * 32) return;
    int n = i >> 5, c = i & 31, hd = c >> 4;
    float v = out1[i] / s1[n * 2 + hd] + b1[c];
    hpost[i] = v > 0.f ? v : (__expf(v) - 1.f);     // ELU(alpha=1)
}

// ---------------- layer 2 projection: h2 = h @ W2 via 8 chained WMMAs -------------
__global__ void k_proj2_wmma(const float* __restrict__ h, const float* __restrict__ W2,
                             float* __restrict__ h2, int nTiles, int N) {
    int lane = threadIdx.x & 31;
    int wave = threadIdx.x >> 5;
    int tile = blockIdx.x * (blockDim.x >> 5) + wave;
    if (tile >= nTiles) return;
    int base = tile * 16;
    int nn   = lane & 15;
    int half = lane >> 4;
    int k0   = half * 2;
    int arow = min(base + nn, N - 1);

    v8f c = {};
#pragma unroll
    for (int kk = 0; kk < 8; ++kk) {               // K = 32 in chunks of 4
        v2f a, b;
        a.x = h[(size_t)arow * 32 + kk * 4 + k0];
        a.y = h[(size_t)arow * 32 + kk * 4 + k0 + 1];
        b.x = (nn < 2) ? W2[(kk * 4 + k0    ) * 2 + nn] : 0.f;
        b.y = (nn < 2) ? W2[(kk * 4 + k0 + 1) * 2 + nn] : 0.f;
        c = __builtin_amdgcn_wmma_f32_16x16x4_f32(false, a, false, b, (short)0, c, false, false);
    }
    if (nn < 2) {
#pragma unroll
        for (int r = 0; r < 8; ++r) {
            int row = base + half * 8 + r;
            if (row < N) h2[(size_t)row * 2 + nn] = c[r];
        }
    }
}

// ---------------- layer 2 edge pass A -------------------------------------------
__global__ void k_l2_max(const long long* __restrict__ ei, long long E, long long ET,
                         const float* __restrict__ h2, const float* __restrict__ a_src2,
                         const float* __restrict__ a_dst2, unsigned* __restrict__ m2) {
    long long t = (long long)blockIdx.x * blockDim.x + threadIdx.x;
    if (t >= ET) return;
    int s, d; edge_sd(ei, E, t, s, d);
    float es = h2[s * 2] * a_src2[0] + h2[s * 2 + 1] * a_src2[1]
             + h2[d * 2] * a_dst2[0] + h2[d * 2 + 1] * a_dst2[1];
    atomicMax(&m2[d], encf(lrelu(es)));
}

// ---------------- layer 2 edge pass B: accumulate into d_out --------------------
__global__ void k_l2_acc(const long long* __restrict__ ei, long long E, long long ET,
                         const float* __restrict__ h2, const float* __restrict__ a_src2,
                         const float* __restrict__ a_dst2, const unsigned* __restrict__ m2,
                         float* __restrict__ dout, float* __restrict__ s2) {
    long long t = (long long)blockIdx.x * blockDim.x + threadIdx.x;
    if (t >= ET) return;
    int s, d; edge_sd(ei, E, t, s, d);
    float h0 = h2[s * 2], h1v = h2[s * 2 + 1];
    float es = h0 * a_src2[0] + h1v * a_src2[1]
             + h2[d * 2] * a_dst2[0] + h2[d * 2 + 1] * a_dst2[1];
    float ex = __expf(lrelu(es) - decf(m2[d]));
    atomicAdd(&dout[d * 2],     ex * h0);
    atomicAdd(&dout[d * 2 + 1], ex * h1v);
    atomicAdd(&s2[d], ex);
}

// ---------------- layer 2 finalize ----------------------------------------------
__global__ void k_fin2(float* __restrict__ dout, const float* __restrict__ s2,
                       const float* __restrict__ b2, int N) {
    int i = blockIdx.x * blockDim.x + threadIdx.x;
    if (i >= N * 2) return;
    dout[i] = dout[i] / s2[i >> 1] + b2[i & 1];
}

extern "C" void kernel_launch(void* const* d_in, const int* in_sizes, int n_in,
                              void* d_out, int out_size, void* d_ws, size_t ws_size,
                              hipStream_t stream) {
    const float*     x      = (const float*)d_in[0];
    const long long* ei     = (const long long*)d_in[1];   // int64 [2,E]
    const float*     W1     = (const float*)d_in[2];
    const float*     a_src1 = (const float*)d_in[3];
    const float*     a_dst1 = (const float*)d_in[4];
    const float*     b1     = (const float*)d_in[5];
    const float*     W2     = (const float*)d_in[6];
    const float*     a_src2 = (const float*)d_in[7];
    const float*     a_dst2 = (const float*)d_in[8];
    const float*     b2     = (const float*)d_in[9];
    float* dout = (float*)d_out;

    const int       N  = in_sizes[0] / 4;        // x is [N,4]
    const long long E  = in_sizes[1] / 2;        // edge_index [2,E]
    const long long ET = E + N;                  // with self-loops

    // carve workspace (floats / uints)
    float* w = (float*)d_ws;
    float*    h1   = w;               w += (size_t)N * 32;   // also reused for post-ELU h
    float*    out1 = w;               w += (size_t)N * 32;
    float*    als1 = w;               w += (size_t)N * 2;
    float*    ald1 = w;               w += (size_t)N * 2;
    float*    s1   = w;               w += (size_t)N * 2;
    float*    h2   = w;               w += (size_t)N * 2;
    float*    s2   = w;               w += (size_t)N;
    unsigned* m1   = (unsigned*)w;    w += (size_t)N * 2;
    unsigned* m2   = (unsigned*)w;    w += (size_t)N;
    (void)ws_size; (void)n_in; (void)out_size;

    const int B = 256;
    const int nTiles = (N + 15) / 16;
    const int wavesPerBlock = B / 32;

    // init accumulators / running maxes / output
    {
        int tot = N * 40;
        k_init<<<(tot + B - 1) / B, B, 0, stream>>>(out1, s1, s2, dout, m1, m2, N);
    }
    // layer 1
    k_proj1_wmma<<<(nTiles + wavesPerBlock - 1) / wavesPerBlock, B, 0, stream>>>(
        x, W1, a_src1, a_dst1, h1, als1, ald1, nTiles, N);
    {
        long long tot = ET * 2;
        k_l1_max<<<(int)((tot + B - 1) / B), B, 0, stream>>>(ei, E, ET, als1, ald1, m1);
    }
    {
        long long tot = ET * 8;
        k_l1_acc<<<(int)((tot + B - 1) / B), B, 0, stream>>>(ei, E, ET, als1, ald1, m1,
                                                             h1, out1, s1);
    }
    k_fin1<<<(N * 32 + B - 1) / B, B, 0, stream>>>(out1, s1, b1, h1, N);  // h1 <- ELU'd features
    // layer 2
    k_proj2_wmma<<<(nTiles + wavesPerBlock - 1) / wavesPerBlock, B, 0, stream>>>(
        h1, W2, h2, nTiles, N);
    k_l2_max<<<(int)((ET + B - 1) / B), B, 0, stream>>>(ei, E, ET, h2, a_src2, a_dst2, m2);
    k_l2_acc<<<(int)((ET + B - 1) / B), B, 0, stream>>>(ei, E, ET, h2, a_src2, a_dst2, m2,
                                                        dout, s2);
    k_fin2<<<(N * 2 + B - 1) / B, B, 0, stream>>>(dout, s2, b2, N);
}